// SetAbstraction_15556371546704
// MI455X (gfx1250) — compile-verified
//
#include <hip/hip_runtime.h>
#include <hip/hip_bf16.h>
#include <math.h>

// ---------------- problem constants ----------------
#define Bc     4
#define Nc     8192
#define CINc   64
#define COUTc  128
#define Kc     16
#define KDc    32          // K * dilation
#define NQc    2048        // N / stride
#define BNQc   (Bc * NQc)  // 8192
#define R2c    0.01f       // radius^2
#define BN_INV 0.999995000037499f   // 1/sqrt(1+1e-5)

typedef __attribute__((ext_vector_type(2))) float v2f;
typedef __attribute__((ext_vector_type(8))) float v8f;

// =====================================================================
// Kernel 1: global furthest-point sampling (one block per batch).
// Per-point running distances live in registers (8 pts / thread).
// Writes idx -> ws, new_p -> d_out section 0.
// =====================================================================
__global__ __launch_bounds__(1024)
void k_fps(const float* __restrict__ p, float* __restrict__ out_newp,
           int* __restrict__ ws_idx) {
  const int b   = blockIdx.x;
  const int tid = threadIdx.x;
  const int lane = tid & 31, wid = tid >> 5;

  __shared__ float sd[32];
  __shared__ int   si[32];
  __shared__ float sxyz[3];
  __shared__ int   sbest;

  const float* pb = p + (size_t)b * Nc * 3;
  float px[8], py[8], pz[8], dd[8];
#pragma unroll
  for (int i = 0; i < 8; ++i) {
    int j = tid * 8 + i;
    px[i] = pb[j * 3 + 0];
    py[i] = pb[j * 3 + 1];
    pz[i] = pb[j * 3 + 2];
    dd[i] = 1e10f;
  }
  int last = 0;
  if (tid == 0) { sxyz[0] = pb[0]; sxyz[1] = pb[1]; sxyz[2] = pb[2]; }
  __syncthreads();
  float lx = sxyz[0], ly = sxyz[1], lz = sxyz[2];

  for (int t = 0; t < NQc; ++t) {
    if (tid == 0) {
      ws_idx[b * NQc + t] = last;
      out_newp[((size_t)b * NQc + t) * 3 + 0] = lx;
      out_newp[((size_t)b * NQc + t) * 3 + 1] = ly;
      out_newp[((size_t)b * NQc + t) * 3 + 2] = lz;
    }
    float bd = -1.0f; int bi = 0;
#pragma unroll
    for (int i = 0; i < 8; ++i) {
      float dx = px[i] - lx, dy = py[i] - ly, dz = pz[i] - lz;
      float d2 = dx * dx + dy * dy + dz * dz;
      dd[i] = fminf(dd[i], d2);
      if (dd[i] > bd) { bd = dd[i]; bi = tid * 8 + i; }
    }
    // wave reduce (argmax, ties -> lower index)
#pragma unroll
    for (int off = 16; off; off >>= 1) {
      float od = __shfl_down(bd, off, 32);
      int   oi = __shfl_down(bi, off, 32);
      if (od > bd || (od == bd && oi < bi)) { bd = od; bi = oi; }
    }
    if (lane == 0) { sd[wid] = bd; si[wid] = bi; }
    __syncthreads();
    if (wid == 0) {
      float wd = sd[lane]; int wi2 = si[lane];
#pragma unroll
      for (int off = 16; off; off >>= 1) {
        float od = __shfl_down(wd, off, 32);
        int   oi = __shfl_down(wi2, off, 32);
        if (od > wd || (od == wd && oi < wi2)) { wd = od; wi2 = oi; }
      }
      if (lane == 0) sbest = wi2;
    }
    __syncthreads();
    last = sbest;
    if (tid == (last >> 3)) {
      int ii = last & 7;
      sxyz[0] = px[ii]; sxyz[1] = py[ii]; sxyz[2] = pz[ii];
    }
    __syncthreads();
    lx = sxyz[0]; ly = sxyz[1]; lz = sxyz[2];
  }
}

// =====================================================================
// Kernel 2: conv1 GEMM  f1 = relu(bn1(W1 @ f))  via V_WMMA_F32_16X16X4_F32.
// One wave per 16x16 tile of [COUT x N]; output stored transposed as
// f1t[b][n][cout] so later per-point feature rows are contiguous.
// A-frag (16x4 f32): lanes 0-15 hold M=lane, K={k0,k0+1}; lanes 16-31 K={k0+2,k0+3}.
// B-frag mirrors A (N=lane%16).
// =====================================================================
__global__ __launch_bounds__(128)
void k_conv1(const float* __restrict__ f, const float* __restrict__ w1,
             const float* __restrict__ g1, const float* __restrict__ b1,
             float* __restrict__ f1t) {
  const int lane = threadIdx.x & 31, wid = threadIdx.x >> 5;
  const int nt = blockIdx.x * 4 + wid;  // 0..511
  const int mt = blockIdx.y;            // 0..7
  const int b  = blockIdx.z;
  const int n0 = nt * 16, m0 = mt * 16;
  const int half = lane >> 4, l = lane & 15;

  const float* fb = f + (size_t)b * CINc * Nc;
  v8f c = {};
#pragma unroll
  for (int k0 = 0; k0 < CINc; k0 += 4) {
    const int ka = k0 + half * 2;
    v2f a, bf;
    a.x  = w1[(m0 + l) * CINc + ka];
    a.y  = w1[(m0 + l) * CINc + ka + 1];
    bf.x = fb[(size_t)ka * Nc + n0 + l];
    bf.y = fb[(size_t)(ka + 1) * Nc + n0 + l];
    c = __builtin_amdgcn_wmma_f32_16x16x4_f32(false, a, false, bf,
                                              (short)0, c, false, false);
  }
#pragma unroll
  for (int vr = 0; vr < 8; ++vr) {
    const int m = m0 + vr + half * 8;   // C/D layout: lanes 0-15 M=vr, 16-31 M=vr+8
    const int n = n0 + l;
    float v = c[vr] * (g1[m] * BN_INV) + b1[m];
    f1t[((size_t)b * Nc + n) * COUTc + m] = fmaxf(v, 0.0f);
  }
}

// =====================================================================
// Kernel 3: ball query, one wave per query point. Forward scan with
// wave32 ballot gives the first KDc in-radius indices in index order
// (exact PointNet++ semantics), padded with the first hit.
// =====================================================================
__global__ __launch_bounds__(256)
void k_ballq(const float* __restrict__ p, const float* __restrict__ newp,
             int* __restrict__ didx) {
  const int lane = threadIdx.x & 31, wid = threadIdx.x >> 5;
  const int q = blockIdx.x * 8 + wid;   // 0..BNQc-1
  const int b = q / NQc;
  const float* pb = p + (size_t)b * Nc * 3;
  const float qx = newp[q * 3 + 0], qy = newp[q * 3 + 1], qz = newp[q * 3 + 2];
  int* dq = didx + (size_t)q * KDc;

  int found = 0, first = -1;
  for (int j0 = 0; j0 < Nc && found < KDc; j0 += 32) {
    __builtin_prefetch(pb + (size_t)(j0 + 32 + lane) * 3, 0, 1);  // global_prefetch_b8
    const int j = j0 + lane;
    const float dx = pb[j * 3 + 0] - qx;
    const float dy = pb[j * 3 + 1] - qy;
    const float dz = pb[j * 3 + 2] - qz;
    const bool pred = (dx * dx + dy * dy + dz * dz) < R2c;
    const unsigned mask = (unsigned)__ballot(pred);
    if (pred) {
      const int pos = found + __popc(mask & ((1u << lane) - 1u));
      if (pos < KDc) dq[pos] = j;
    }
    if (first < 0 && mask) first = j0 + __ffs(mask) - 1;
    found += __popc(mask);
  }
  if (first < 0) { first = 0; if (lane == 0) dq[0] = 0; found = 1; }
  for (int s = found + lane; s < KDc; s += 32) dq[s] = first;  // pad
}

// =====================================================================
// Kernel 4: per-neighborhood (one wave each): FPS 32->16, KDE offsets,
// 1-NN reassignment. KDE MLP weights staged in LDS. Outputs dp_cl and
// the global feature-gather index per kernel point.
// =====================================================================
__global__ __launch_bounds__(256)
void k_neigh(const float* __restrict__ p, const float* __restrict__ newp,
             const int* __restrict__ didx,
             const float* __restrict__ kw1, const float* __restrict__ kb1,
             const float* __restrict__ kw2, const float* __restrict__ kb2,
             const float* __restrict__ kw3, const float* __restrict__ kb3,
             int* __restrict__ nng, float* __restrict__ dpcl) {
  const int lane = threadIdx.x & 31, wid = threadIdx.x >> 5;
  const int g = blockIdx.x * 8 + wid;   // neighborhood id
  const int b = g / NQc;

  __shared__ float s[707];  // kw1[96] kb1[32] kw2[512] kb2[16] kw3[48] kb3[3]
  float* sk1  = s;        float* sk1b = s + 96;
  float* sk2  = s + 128;  float* sk2b = s + 640;
  float* sk3  = s + 656;  float* sk3b = s + 704;
  for (int i = threadIdx.x; i < 96;  i += 256) sk1[i]  = kw1[i];
  for (int i = threadIdx.x; i < 32;  i += 256) sk1b[i] = kb1[i];
  for (int i = threadIdx.x; i < 512; i += 256) sk2[i]  = kw2[i];
  for (int i = threadIdx.x; i < 16;  i += 256) sk2b[i] = kb2[i];
  for (int i = threadIdx.x; i < 48;  i += 256) sk3[i]  = kw3[i];
  for (int i = threadIdx.x; i < 3;   i += 256) sk3b[i] = kb3[i];
  __syncthreads();

  const int j = didx[(size_t)g * KDc + lane];
  const float* pb = p + (size_t)b * Nc * 3;
  const float qx = newp[g * 3 + 0], qy = newp[g * 3 + 1], qz = newp[g * 3 + 2];
  const float x = pb[j * 3 + 0] - qx, y = pb[j * 3 + 1] - qy, z = pb[j * 3 + 2] - qz;

  // ---- FPS over 32 dilated points selecting Kc kernel points ----
  float d = 1e10f; int last = 0, sel = 0;
  for (int t = 0; t < Kc; ++t) {
    if (lane == t) sel = last;
    const float lxx = __shfl(x, last, 32), lyy = __shfl(y, last, 32), lzz = __shfl(z, last, 32);
    const float dx = x - lxx, dy = y - lyy, dz = z - lzz;
    d = fminf(d, dx * dx + dy * dy + dz * dz);
    float bd = d; int bi = lane;
#pragma unroll
    for (int off = 16; off; off >>= 1) {
      const float od = __shfl_xor(bd, off, 32);
      const int   oi = __shfl_xor(bi, off, 32);
      if (od > bd || (od == bd && oi < bi)) { bd = od; bi = oi; }
    }
    last = bi;
  }
  // kernel point for lane i (<16): dilated point 'sel'
  const float kx = __shfl(x, sel, 32), ky = __shfl(y, sel, 32), kz = __shfl(z, sel, 32);
  const int   gj = __shfl(j, sel, 32);

  // ---- KDE offsets (lanes 0..15 meaningful) ----
  float pe[32];
#pragma unroll
  for (int o = 0; o < 32; ++o) pe[o] = 0.0f;
  float kde = 0.0f;
  const float coef   = 0.39894225f;         // 1/sqrt(2*pi*sigma^2 + eps)
  const float invden = 1.0f / 2.000001f;    // 1/(2*sigma^2 + eps)
  for (int jj = 0; jj < Kc; ++jj) {
    const float rx = kx - __shfl(kx, jj, 32);
    const float ry = ky - __shfl(ky, jj, 32);
    const float rz = kz - __shfl(kz, jj, 32);
    kde += coef * expf(-(rx * rx + ry * ry + rz * rz) * invden);
#pragma unroll
    for (int o = 0; o < 32; ++o) {
      const float v = sk1[o * 3] * rx + sk1[o * 3 + 1] * ry + sk1[o * 3 + 2] * rz + sk1b[o];
      pe[o] += fmaxf(v, 0.0f);
    }
  }
#pragma unroll
  for (int o = 0; o < 32; ++o) pe[o] *= (1.0f / 16.0f);
  float km = kde;
#pragma unroll
  for (int off = 8; off; off >>= 1) km = fmaxf(km, __shfl_xor(km, off, 32));  // max over 16 kernel pts
  kde = kde / (km + 1e-6f);

  float h[16];
#pragma unroll
  for (int hh = 0; hh < 16; ++hh) {
    float v = sk2b[hh];
#pragma unroll
    for (int o = 0; o < 32; ++o) v += sk2[hh * 32 + o] * (pe[o] * kde);
    h[hh] = fmaxf(v, 0.0f);
  }
  float dox = sk3b[0], doy = sk3b[1], doz = sk3b[2];
#pragma unroll
  for (int hh = 0; hh < 16; ++hh) {
    dox += sk3[0 * 16 + hh] * h[hh];
    doy += sk3[1 * 16 + hh] * h[hh];
    doz += sk3[2 * 16 + hh] * h[hh];
  }
  const float defx = kx + dox, defy = ky + doy, defz = kz + doz;

  // ---- 1-NN of deformed point among kernel points (first-min) ----
  float best = 1e30f; int nn = 0;
  for (int jj = 0; jj < Kc; ++jj) {
    const float rx = defx - __shfl(kx, jj, 32);
    const float ry = defy - __shfl(ky, jj, 32);
    const float rz = defz - __shfl(kz, jj, 32);
    const float d2 = rx * rx + ry * ry + rz * rz;
    if (d2 < best) { best = d2; nn = jj; }
  }
  const float nx = __shfl(kx, nn, 32), ny = __shfl(ky, nn, 32), nz = __shfl(kz, nn, 32);
  const int gnn  = __shfl(gj, nn, 32);

  if (lane < Kc) {
    dpcl[((size_t)g * Kc + lane) * 3 + 0] = nx - defx;
    dpcl[((size_t)g * Kc + lane) * 3 + 1] = ny - defy;
    dpcl[((size_t)g * Kc + lane) * 3 + 2] = nz - defz;
    nng[(size_t)g * Kc + lane] = gnn;
  }
}

// =====================================================================
// Kernel 5: fused head. One wave = one 16(cout)x16(k) tile of ONE
// neighborhood (K==16 => tile columns are exactly the softmax axis).
// WMMA wconv GEMM (K=128 contraction) -> bn -> softmax over k (width-16
// shfl) -> pe_out (K=3 FMA) -> (pe+fj)*mod -> row-max -> f_out.
// =====================================================================
__global__ __launch_bounds__(256)
void k_head(const float* __restrict__ f1t, const int* __restrict__ nng,
            const float* __restrict__ dpcl,
            const float* __restrict__ wc, const float* __restrict__ gw,
            const float* __restrict__ bw,
            const float* __restrict__ w2c, const float* __restrict__ g2,
            const float* __restrict__ b2,
            float* __restrict__ out_f, float* __restrict__ out_pe) {
  const int lane = threadIdx.x & 31, wid = threadIdx.x >> 5;
  const int g  = blockIdx.x * 8 + wid;   // neighborhood
  const int mt = blockIdx.y;             // cout tile 0..7
  const int b  = g / NQc, qi = g % NQc;
  const int half = lane >> 4, l = lane & 15;
  const int m0 = mt * 16;

  const int jcol = nng[(size_t)g * Kc + l];               // feature source of column k=l
  const float* frow = f1t + ((size_t)b * Nc + jcol) * COUTc;

  v8f c = {};
#pragma unroll
  for (int k0 = 0; k0 < COUTc; k0 += 4) {
    const int ka = k0 + half * 2;
    v2f a, bf;
    a.x  = wc[(m0 + l) * COUTc + ka];
    a.y  = wc[(m0 + l) * COUTc + ka + 1];
    bf.x = frow[ka];
    bf.y = frow[ka + 1];
    c = __builtin_amdgcn_wmma_f32_16x16x4_f32(false, a, false, bf,
                                              (short)0, c, false, false);
  }

  const float d0 = dpcl[((size_t)g * Kc + l) * 3 + 0];
  const float d1 = dpcl[((size_t)g * Kc + l) * 3 + 1];
  const float d2 = dpcl[((size_t)g * Kc + l) * 3 + 2];

#pragma unroll
  for (int vr = 0; vr < 8; ++vr) {
    const int m = m0 + vr + half * 8;
    // bn on logits
    const float lg = c[vr] * (gw[m] * BN_INV) + bw[m];
    // softmax over k: reduce across the 16-lane group (offsets < 16 keep halves separate)
    float mx = lg;
#pragma unroll
    for (int off = 8; off; off >>= 1) mx = fmaxf(mx, __shfl_xor(mx, off, 32));
    const float e = expf(lg - mx);
    float sum = e;
#pragma unroll
    for (int off = 8; off; off >>= 1) sum += __shfl_xor(sum, off, 32);
    const float mod = e / sum;
    // pe_out = relu(bn2(conv2_w @ dp_cl))
    float peo = (w2c[m * 3] * d0 + w2c[m * 3 + 1] * d1 + w2c[m * 3 + 2] * d2)
                * (g2[m] * BN_INV) + b2[m];
    peo = fmaxf(peo, 0.0f);
    out_pe[(((size_t)b * COUTc + m) * NQc + qi) * Kc + l] = peo;
    // modulated combine + max over k
    const float comb = (peo + frow[m]) * mod;
    float rm = comb;
#pragma unroll
    for (int off = 8; off; off >>= 1) rm = fmaxf(rm, __shfl_xor(rm, off, 32));
    if (l == 0) out_f[((size_t)b * COUTc + m) * NQc + qi] = rm;
  }
}

// =====================================================================
// launch
// =====================================================================
extern "C" void kernel_launch(void* const* d_in, const int* in_sizes, int n_in,
                              void* d_out, int out_size, void* d_ws, size_t ws_size,
                              hipStream_t stream) {
  const float* p       = (const float*)d_in[0];
  const float* f       = (const float*)d_in[1];
  // d_in[2] = pe (unused dummy)
  const float* conv1_w = (const float*)d_in[3];
  const float* bn1_g   = (const float*)d_in[4];
  const float* bn1_b   = (const float*)d_in[5];
  const float* conv2_w = (const float*)d_in[6];
  const float* bn2_g   = (const float*)d_in[7];
  const float* bn2_b   = (const float*)d_in[8];
  const float* wconv_w = (const float*)d_in[9];
  const float* bnw_g   = (const float*)d_in[10];
  const float* bnw_b   = (const float*)d_in[11];
  const float* kde_w1  = (const float*)d_in[12];
  const float* kde_b1  = (const float*)d_in[13];
  const float* kde_w2  = (const float*)d_in[14];
  const float* kde_b2  = (const float*)d_in[15];
  const float* kde_w3  = (const float*)d_in[16];
  const float* kde_b3  = (const float*)d_in[17];

  // output sections (tuple return order: new_p, f_out, pe_out)
  float* out_newp = (float*)d_out;                             // B*NQ*3   = 24576
  float* out_f    = out_newp + (size_t)Bc * NQc * 3;           // B*COUT*NQ = 1048576
  float* out_pe   = out_f + (size_t)Bc * COUTc * NQc;          // B*COUT*NQ*K

  // workspace layout (~20 MB)
  int*   ws_idx  = (int*)d_ws;                                 // B*NQ
  int*   ws_didx = ws_idx + (size_t)Bc * NQc;                  // B*NQ*KD
  int*   ws_nng  = ws_didx + (size_t)Bc * NQc * KDc;           // BNQ*K
  float* ws_f1t  = (float*)(ws_nng + (size_t)BNQc * Kc);       // B*N*COUT
  float* ws_dpcl = ws_f1t + (size_t)Bc * Nc * COUTc;           // BNQ*K*3

  k_fps  <<<Bc, 1024, 0, stream>>>(p, out_newp, ws_idx);
  k_conv1<<<dim3(128, 8, Bc), 128, 0, stream>>>(f, conv1_w, bn1_g, bn1_b, ws_f1t);
  k_ballq<<<BNQc / 8, 256, 0, stream>>>(p, out_newp, ws_didx);
  k_neigh<<<BNQc / 8, 256, 0, stream>>>(p, out_newp, ws_didx,
                                        kde_w1, kde_b1, kde_w2, kde_b2, kde_w3, kde_b3,
                                        ws_nng, ws_dpcl);
  k_head <<<dim3(BNQc / 8, 8), 256, 0, stream>>>(ws_f1t, ws_nng, ws_dpcl,
                                                 wconv_w, bnw_g, bnw_b,
                                                 conv2_w, bn2_g, bn2_b,
                                                 out_f, out_pe);
}